// Ensemble_SRN_61108794687855
// MI455X (gfx1250) — compile-verified
//
#include <hip/hip_runtime.h>

typedef __attribute__((ext_vector_type(2))) float v2f;
typedef __attribute__((ext_vector_type(4))) float v4f;
typedef __attribute__((ext_vector_type(8))) float v8f;

#define NPTS 524288
#define NEXP 8
#define HID 64
#define TILE 16
#define WPB 8                       // waves per block (256 threads)
#define PERM_TOTAL (NPTS + NEXP*TILE)

// workspace int layout
#define WS_COUNT 0                  // [0..7]  per-expert counts
#define WS_OFF   8                  // [8..16] padded offsets, [16] = padded total
#define WS_CUR   17                 // [17..24] scatter cursors
#define WS_PERM  32                 // [32..]   perm array, PERM_TOTAL ints

// LDS layout (floats)
#define L_W1T 0                                   // NEXP*HID*4  : [e][c][{w0,w1,w2,b1}]
#define L_W2T (L_W1T + NEXP*HID*4)                // NEXP*HID*HID: [e][n][k] (transposed)
#define L_B2  (L_W2T + NEXP*HID*HID)              // NEXP*HID
#define L_W3  (L_B2  + NEXP*HID)                  // NEXP*HID
#define L_Y   (L_W3  + NEXP*HID)                  // WPB*16
#define L_TOT (L_Y   + WPB*16)
#define SMEM_BYTES (L_TOT * 4)

__device__ __forceinline__ int route_idx(const float* __restrict__ x, int t) {
  float px = x[3*t+0], py = x[3*t+1], pz = x[3*t+2];
  float u0 = fminf(fmaxf((px + 1.0f) * 0.5f, 0.0f), 0.99f);
  float u1 = fminf(fmaxf((py + 1.0f) * 0.5f, 0.0f), 0.99f);
  float u2 = fminf(fmaxf((pz + 1.0f) * 0.5f, 0.0f), 0.99f);
  int g0 = (int)(u0 * 2.0f);
  int g1 = (int)(u1 * 2.0f);
  int g2 = (int)(u2 * 2.0f);
  return g0 + g1 * 2 + g2 * 4;
}

__global__ __launch_bounds__(256) void k_init(int* __restrict__ ws) {
  int t = blockIdx.x * blockDim.x + threadIdx.x;
  if (t < 32) ws[t] = 0;
  if (t < PERM_TOTAL) ws[WS_PERM + t] = -1;
}

__global__ __launch_bounds__(256) void k_count(const float* __restrict__ x,
                                               int* __restrict__ ws) {
  int t = blockIdx.x * blockDim.x + threadIdx.x;
  if (t >= NPTS) return;
  atomicAdd(&ws[WS_COUNT + route_idx(x, t)], 1);
}

__global__ void k_offsets(int* __restrict__ ws) {
  if (threadIdx.x == 0 && blockIdx.x == 0) {
    int off = 0;
    for (int e = 0; e < NEXP; ++e) {
      ws[WS_OFF + e] = off;
      off += ((ws[WS_COUNT + e] + TILE - 1) / TILE) * TILE;   // pad to tile
    }
    ws[WS_OFF + NEXP] = off;
  }
}

__global__ __launch_bounds__(256) void k_scatter(const float* __restrict__ x,
                                                 int* __restrict__ ws) {
  int t = blockIdx.x * blockDim.x + threadIdx.x;
  if (t >= NPTS) return;
  int e = route_idx(x, t);
  int pos = atomicAdd(&ws[WS_CUR + e], 1);
  ws[WS_PERM + ws[WS_OFF + e] + pos] = t;
}

__global__ __launch_bounds__(256) void k_mlp(
    const float* __restrict__ x,
    const float* __restrict__ emin, const float* __restrict__ emax,
    const float* __restrict__ W1, const float* __restrict__ b1,
    const float* __restrict__ W2, const float* __restrict__ b2,
    const float* __restrict__ W3, const float* __restrict__ b3,
    const int* __restrict__ ws, float* __restrict__ out) {
  extern __shared__ float smem[];
  const int tid = threadIdx.x;

  // ---- cooperative LDS staging: all 8 experts' weights ----
  // W2 transposed: smem[L_W2T + e*4096 + n*64 + k] = W2[e][k][n]
  for (int i = tid; i < NEXP * HID * HID; i += 256) {
    int e = i >> 12, rem = i & 4095, k = rem >> 6, n = rem & 63;
    smem[L_W2T + ((e << 12) | (n << 6) | k)] = W2[i];
  }
  // W1 fused with b1: smem[L_W1T + (e*64+c)*4 + {0,1,2,3}] = {W1[e][0][c],W1[e][1][c],W1[e][2][c],b1[e][c]}
  for (int i = tid; i < NEXP * HID; i += 256) {
    int e = i >> 6, c = i & 63;
    float* d = &smem[L_W1T + i * 4];
    d[0] = W1[(e * 3 + 0) * HID + c];
    d[1] = W1[(e * 3 + 1) * HID + c];
    d[2] = W1[(e * 3 + 2) * HID + c];
    d[3] = b1[i];
    smem[L_B2 + i] = b2[i];
    smem[L_W3 + i] = W3[i];
  }
  __syncthreads();

  const int lane  = tid & 31;
  const int wave  = tid >> 5;
  const int halfk = lane >> 4;        // K-half for A/B fragments
  const int m     = lane & 15;        // row for A, column for B/C/D

  const int tileIdx = blockIdx.x * WPB + wave;
  const int base    = tileIdx * TILE;
  const int total   = ws[WS_OFF + NEXP];
  if (base >= total) return;          // wave-uniform exit (no barriers after)

  // which expert owns this tile
  int e = 0;
#pragma unroll
  for (int i = 1; i < NEXP; ++i)
    if (base >= ws[WS_OFF + i]) e = i;

  // gather this row's point (lanes L and L+16 share row m)
  const int pid = ws[WS_PERM + base + m];
  float xn0 = 0.f, xn1 = 0.f, xn2 = 0.f;
  if (pid >= 0) {
    float mn0 = emin[e * 3 + 0], mn1 = emin[e * 3 + 1], mn2 = emin[e * 3 + 2];
    float s0 = 2.0f / (emax[e * 3 + 0] - mn0);
    float s1 = 2.0f / (emax[e * 3 + 1] - mn1);
    float s2 = 2.0f / (emax[e * 3 + 2] - mn2);
    xn0 = (x[pid * 3 + 0] - mn0) * s0 - 1.0f;
    xn1 = (x[pid * 3 + 1] - mn1) * s1 - 1.0f;
    xn2 = (x[pid * 3 + 2] - mn2) * s2 - 1.0f;
  }

  // ---- layer 1 on VALU, producing h1 directly in WMMA A-fragment layout ----
  // k-step s covers k = 4s..4s+3; this lane holds k = 4s+2*halfk and 4s+2*halfk+1
  float a0[16], a1[16];
  const float* w1t = &smem[L_W1T + e * HID * 4];
#pragma unroll
  for (int s = 0; s < 16; ++s) {
    int c0 = 4 * s + 2 * halfk;
    v4f w0 = *(const v4f*)&w1t[c0 * 4];
    v4f w1 = *(const v4f*)&w1t[(c0 + 1) * 4];
    a0[s] = fmaxf(w0.w + xn0 * w0.x + xn1 * w0.y + xn2 * w0.z, 0.0f);
    a1[s] = fmaxf(w1.w + xn0 * w1.x + xn1 * w1.y + xn2 * w1.z, 0.0f);
  }

  // ---- layer 2: 16x64 @ 64x64 via 16 k-steps x 4 n-tiles of f32 WMMA ----
  v8f acc[4];
#pragma unroll
  for (int j = 0; j < 4; ++j) {
    float bb = smem[L_B2 + e * HID + j * 16 + m];
    acc[j] = (v8f){bb, bb, bb, bb, bb, bb, bb, bb};
  }
  const float* w2t = &smem[L_W2T + e * HID * HID];
#pragma unroll
  for (int s = 0; s < 16; ++s) {
    v2f A;
    A.x = a0[s];
    A.y = a1[s];
#pragma unroll
    for (int j = 0; j < 4; ++j) {
      int n = j * 16 + m;
      v2f B = *(const v2f*)&w2t[n * HID + 4 * s + 2 * halfk];
      acc[j] = __builtin_amdgcn_wmma_f32_16x16x4_f32(
          false, A, false, B, (short)0, acc[j], false, false);
    }
  }

  // ---- layer 3: relu + dot with W3, reduce across 16 lanes per half ----
  float partial[8];
#pragma unroll
  for (int r = 0; r < 8; ++r) {
    float sum = 0.0f;
#pragma unroll
    for (int j = 0; j < 4; ++j)
      sum += fmaxf(acc[j][r], 0.0f) * smem[L_W3 + e * HID + j * 16 + m];
    partial[r] = sum;
  }
#pragma unroll
  for (int off = 1; off < 16; off <<= 1) {
#pragma unroll
    for (int r = 0; r < 8; ++r)
      partial[r] += __shfl_xor(partial[r], off, 16);
  }
  // lane m==r in each half owns row halfk*8+r; stage through LDS strip
#pragma unroll
  for (int r = 0; r < 8; ++r)
    if (m == r) smem[L_Y + wave * 16 + halfk * 8 + r] = partial[r];

  if (halfk == 0 && pid >= 0)
    out[pid] = smem[L_Y + wave * 16 + m] + b3[e];
}

extern "C" void kernel_launch(void* const* d_in, const int* in_sizes, int n_in,
                              void* d_out, int out_size, void* d_ws, size_t ws_size,
                              hipStream_t stream) {
  const float* x    = (const float*)d_in[0];
  const float* emin = (const float*)d_in[1];
  const float* emax = (const float*)d_in[2];
  const float* W1   = (const float*)d_in[3];
  const float* b1   = (const float*)d_in[4];
  const float* W2   = (const float*)d_in[5];
  const float* b2   = (const float*)d_in[6];
  const float* W3   = (const float*)d_in[7];
  const float* b3   = (const float*)d_in[8];
  float* out = (float*)d_out;
  int* ws = (int*)d_ws;

  int initBlocks = (PERM_TOTAL + 255) / 256;
  int ptBlocks   = (NPTS + 255) / 256;
  int mlpBlocks  = (PERM_TOTAL / TILE + WPB - 1) / WPB;

  k_init<<<initBlocks, 256, 0, stream>>>(ws);
  k_count<<<ptBlocks, 256, 0, stream>>>(x, ws);
  k_offsets<<<1, 32, 0, stream>>>(ws);
  k_scatter<<<ptBlocks, 256, 0, stream>>>(x, ws);
  k_mlp<<<mlpBlocks, 256, SMEM_BYTES, stream>>>(x, emin, emax, W1, b1, W2, b2,
                                                W3, b3, ws, out);
}